// relative_pos_embedding_5746666242386
// MI455X (gfx1250) — compile-verified
//
#include <hip/hip_runtime.h>
#include <cmath>

// ---------------------------------------------------------------------------
// out[b,h,t,r] = sum_d q[b,h,t,d] * emb[h, indices[r,t], d]
// B=H=8, T=R=512, D=64, P=2T-1=1023, indices stride 1490.
//
// Per (h,t):  C(8x512) = Q(8x64) x G_t^T(64x512),  G_t[r,:] = emb[h, idx[r,t], :]
// -> V_WMMA_F32_16X16X4_F32, K=64 as 2 interleaved chains of 8 WMMAs.
// emb[h] (256KB) staged in LDS via TDM with +4 DWORD row padding (stride
// 68 floats) so the 16-row ds_load_b64 gather is bank-conflict free.
// Index gather software-pipelined by 1 tile; all global addressing 32-bit.
// ---------------------------------------------------------------------------

typedef float v2f __attribute__((ext_vector_type(2)));
typedef float v8f __attribute__((ext_vector_type(8)));
typedef unsigned int v4u __attribute__((ext_vector_type(4)));
typedef int v8i __attribute__((ext_vector_type(8)));
typedef int v4i __attribute__((ext_vector_type(4)));

#if __has_builtin(__builtin_amdgcn_tensor_load_to_lds)
#define USE_TDM 1
#else
#define USE_TDM 0
#endif

constexpr int B = 8, H = 8, T = 512, D = 64, R = 512, P = 1023;
constexpr int ROW = 68;                       // padded LDS row stride (floats)
constexpr int LDS_BYTES = P * ROW * 4;        // 278,256 B <= 320 KB
constexpr int TT = 16;                        // t-values per workgroup

__global__ __launch_bounds__(256)
void relpos_wmma_kernel(const float* __restrict__ q,
                        const float* __restrict__ emb,
                        const int* __restrict__ indices,
                        float* __restrict__ out,
                        int istride) {
  extern __shared__ float lds[];              // emb[h], padded rows
  const int h    = blockIdx.y;
  const int t0   = blockIdx.x * TT;
  const int tid  = threadIdx.x;
  const int lane = tid & 31;
  const int wave = tid >> 5;
  const int m    = lane & 15;                 // A-row (=b) / B-col (=n within tile)
  const int koff = (lane >> 4) * 2;           // K offset inside a 4-slice: 0 or 2

  // ---- Stage emb[h] -> LDS --------------------------------------------------
#if USE_TDM
  // Tensor Data Mover: 2D tile 64 x 1023 of 4B elements, LDS padding of
  // 4 DWORDs after every 64 DWORDs stored (pad_interval=5, pad_amount=3)
  // -> LDS row stride 68 floats. Dynamic LDS starts at offset 0 (no static LDS).
  if (tid < 32) {
    unsigned long long ga = (unsigned long long)(const void*)(emb + (size_t)h * P * D);
    v4u g0 = { 1u,                                   // count=1 (valid descriptor)
               0u,                                   // lds_addr = 0
               (unsigned)(ga & 0xFFFFFFFFu),         // global_addr[31:0]
               (unsigned)((ga >> 32) & 0x1FFFFFFu) | (2u << 30) };  // addr[56:32] | type=2
    v8i g1 = { (2 << 16) | (1 << 20) | (5 << 22) | (3 << 25), // data_size=4B, pad_en, intv=64dw, amt=4dw
               (D << 16),                            // tensor_dim0 = 64 (low 16 in [31:16])
               (P << 16),                            // tensor_dim0 hi=0 | tensor_dim1 = 1023
               (D << 16),                            // tensor_dim1 hi=0 | tile_dim0 = 64
               P,                                    // tile_dim1 = 1023, tile_dim2 = 0
               D,                                    // tensor_dim0_stride = 64
               0, 0 };                               // stride hi / dim1_stride = 0
    v4i gz4 = { 0, 0, 0, 0 };                        // group2 (unused: 2D tile)
    v4i gz4b = { 0, 0, 0, 0 };                       // group3 (unused: 2D tile)
    v8i gz8 = { 0, 0, 0, 0, 0, 0, 0, 0 };            // extra arg (6-arg toolchain)
    __builtin_amdgcn_tensor_load_to_lds(g0, g1, gz4, gz4b, gz8, 0);
    __builtin_amdgcn_s_wait_tensorcnt(0);
  }
#else
  // Fallback: cooperative float4 fill with the same 68-float row stride.
  for (int v = tid; v < P * (D / 4); v += 256) {
    const int row = v >> 4;
    const int c4  = (v & 15) << 2;
    const float4 val = *(const float4*)(emb + ((size_t)h * P + row) * D + c4);
    *(float4*)(lds + row * ROW + c4) = val;
  }
#endif
  __syncthreads();

  const v2f z2 = { 0.0f, 0.0f };
  const unsigned istep = 16u * (unsigned)istride;

  // 8 waves x 2 t-values = 16 t per block
  for (int tt = 0; tt < 2; ++tt) {
    const int t = t0 + wave * 2 + tt;

    // ---- A fragments: Q[b=m, h, t, :], 16 K-slices of 2 floats per lane ----
    // Layout (16x4 fp32 A): lanes 0-15 M=0..15 K={0,1}; lanes 16-31 K={2,3}.
    const int bq = (m < 8) ? m : 0;             // clamp, then select zero
    const unsigned qidx = ((((unsigned)bq * H + h) * T + t) * D) + koff;  // 32-bit
    v2f a[16];
#pragma unroll
    for (int kk = 0; kk < 16; ++kk) {
      v2f av = *(const v2f*)(q + qidx + kk * 4);
      a[kk] = (m < 8) ? av : z2;
    }

    // Per-lane gather-row index, software-pipelined by one tile.
    const int* icol = indices + t;
    unsigned iofs = (unsigned)m * (unsigned)istride;   // row m, column t
    int p = icol[iofs];                                // tile nt=0

    // 32-bit output element base for this (h,t,lane): b stride added per j.
    const unsigned obase = (((unsigned)h * T + (unsigned)t) * R) + (unsigned)m;

#pragma unroll 1
    for (int nt = 0; nt < 32; ++nt) {
      iofs += istep;
      const int pn = icol[iofs];            // prefetch next tile's row (<=527, in-bounds)

      // B fragments: G_t[r, :] from LDS (ds_load_2addr_b64, conflict-free).
      const float* grow = lds + p * ROW + koff;
      v2f bf[16];
#pragma unroll
      for (int kk = 0; kk < 16; ++kk) bf[kk] = *(const v2f*)(grow + kk * 4);

      // Two independent accumulation chains (even/odd K-slices) for ILP.
      v8f c0 = { 0, 0, 0, 0, 0, 0, 0, 0 };
      v8f c1 = { 0, 0, 0, 0, 0, 0, 0, 0 };
#pragma unroll
      for (int kk = 0; kk < 8; ++kk) {
        c0 = __builtin_amdgcn_wmma_f32_16x16x4_f32(false, a[2 * kk],     false, bf[2 * kk],
                                                   (short)0, c0, false, false);
        c1 = __builtin_amdgcn_wmma_f32_16x16x4_f32(false, a[2 * kk + 1], false, bf[2 * kk + 1],
                                                   (short)0, c1, false, false);
      }
      const v8f c = c0 + c1;

      // D layout: VGPR j = rows M=j (lanes 0-15) / M=j+8 (lanes 16-31).
      // Valid outputs: lanes 0-15, b = j. Output never re-read -> NT stores.
      if (lane < 16) {
        const unsigned ob = obase + (unsigned)nt * 16u;
#pragma unroll
        for (int j = 0; j < 8; ++j) {
          __builtin_nontemporal_store(c[j], out + ob + (unsigned)j * (unsigned)(H * T * R));
        }
      }
      p = pn;
    }
  }
}

extern "C" void kernel_launch(void* const* d_in, const int* in_sizes, int n_in,
                              void* d_out, int out_size, void* d_ws, size_t ws_size,
                              hipStream_t stream) {
  const float* q       = (const float*)d_in[0];
  const float* emb     = (const float*)d_in[1];
  const int*   indices = (const int*)d_in[2];
  float*       out     = (float*)d_out;

  int istride = 1490;                          // indices is square (1490x1490)
  if (n_in > 2 && in_sizes[2] > 0) {
    int s = (int)(std::sqrt((double)in_sizes[2]) + 0.5);
    if (s > 0 && s * s == in_sizes[2]) istride = s;
  }

  (void)hipFuncSetAttribute(reinterpret_cast<const void*>(&relpos_wmma_kernel),
                            hipFuncAttributeMaxDynamicSharedMemorySize, LDS_BYTES);

  relpos_wmma_kernel<<<dim3(T / TT, H), 256, LDS_BYTES, stream>>>(
      q, emb, indices, out, istride);
}